// Encoder_6476810682593
// MI455X (gfx1250) — compile-verified
//
#include <hip/hip_runtime.h>
#include <hip/hip_bf16.h>

// ---------------------------------------------------------------------------
// GCN3D-style point-cloud encoder for MI455X (gfx1250, wave32).
//  - kNN scans: double-buffered async global->LDS tile staging (ASYNCcnt),
//    register-cached reject threshold, scratch only on real insertions
//  - the two dense feature GEMMs use V_WMMA_F32_16X16X4_F32 (fp32 matrix pipe)
//  - per-neighbor theta/max/sum reductions are VALU with shfl / LDS broadcast
// ---------------------------------------------------------------------------

typedef float v2f __attribute__((ext_vector_type(2)));
typedef float v8f __attribute__((ext_vector_type(8)));

static constexpr int BS = 8;
static constexpr int N0 = 2048;   // points, stage 0/1
static constexpr int N1 = 256;    // points after pool 1
static constexpr int N2 = 16;     // points after pool 2
static constexpr int NB = 20;     // neighbors for operators
static constexpr int SUP = 7;     // support_num
static constexpr int K0 = 32, C1 = 64, C2 = 256;
static constexpr int STRIDE1 = 997;  // odd -> bijection mod 2048
static constexpr int STRIDE2 = 163;  // odd -> bijection mod 256

// ---------------- workspace layout (bytes) ----------------
static constexpr size_t AL = 256;
static constexpr size_t align_up(size_t x) { return (x + AL - 1) & ~(AL - 1); }
static constexpr size_t OFF_NI20  = 0;
static constexpr size_t OFF_FM0   = align_up(OFF_NI20  + (size_t)BS*N0*NB*4);
static constexpr size_t OFF_FEAT1 = align_up(OFF_FM0   + (size_t)BS*N0*K0*4);
static constexpr size_t OFF_FM1   = align_up(OFF_FEAT1 + (size_t)BS*N0*(SUP+1)*C1*4);
static constexpr size_t OFF_NI8   = align_up(OFF_FM1   + (size_t)BS*N0*C1*4);
static constexpr size_t OFF_V1    = align_up(OFF_NI8   + (size_t)BS*N1*8*4);
static constexpr size_t OFF_FM1P  = align_up(OFF_V1    + (size_t)BS*N1*3*4);
static constexpr size_t OFF_NI20B = align_up(OFF_FM1P  + (size_t)BS*N1*C1*4);
static constexpr size_t OFF_FEAT2 = align_up(OFF_NI20B + (size_t)BS*N1*NB*4);
static constexpr size_t OFF_FM2   = align_up(OFF_FEAT2 + (size_t)BS*N1*(SUP+1)*C2*4);
static constexpr size_t OFF_NI16  = align_up(OFF_FM2   + (size_t)BS*N1*C2*4);

// ---------------------------------------------------------------------------
// kNN: per-thread query against all V points. Candidate tiles are staged into
// LDS with async global->LDS loads (double buffered, overlapped with the scan).
// Top-(K+1) insertion list, drop self (slot 0).  V is a power of two and a
// multiple of KNN_TILE; blockDim.x == KNN_TILE.  Query rows = (q*qstride)&(V-1).
// ---------------------------------------------------------------------------
static constexpr int KNN_TILE = 128;

template <int K>
__global__ void knn_kernel(const float* __restrict__ pts, int V, int Q,
                           int qstride, int* __restrict__ out) {
  constexpr int KP1 = K + 1;
  const int b = blockIdx.y;
  const int q = blockIdx.x * blockDim.x + threadIdx.x;
  const bool active = q < Q;
  const int row = (q * qstride) & (V - 1);
  const float* P = pts + (size_t)b * V * 3;

  float qx = 0.f, qy = 0.f, qz = 0.f, qn = 0.f;
  if (active) {
    qx = P[row * 3 + 0]; qy = P[row * 3 + 1]; qz = P[row * 3 + 2];
    qn = qx * qx + qy * qy + qz * qz;
  }

  float bd[KP1]; int bi[KP1];
#pragma unroll
  for (int t = 0; t < KP1; ++t) { bd[t] = 3.0e38f; bi[t] = 0; }
  float worst = 3.0e38f;          // == bd[KP1-1], register-resident reject test

  __shared__ float tile[2][KNN_TILE * 3];

  // Issue KNN_TILE*3 async word copies (3 per thread) for one candidate tile.
  auto issue_tile = [&](int base, int buf) {
#pragma unroll
    for (int w = 0; w < 3; ++w) {
      const int t = threadIdx.x + w * KNN_TILE;
      const unsigned lds_off = (unsigned)(size_t)&tile[buf][t];
      const float* g = &P[base * 3 + t];
      asm volatile("global_load_async_to_lds_b32 %0, %1, off"
                   :: "v"(lds_off), "v"(g) : "memory");
    }
  };

  issue_tile(0, 0);
  for (int base = 0; base < V; base += KNN_TILE) {
    const int buf = (base / KNN_TILE) & 1;
    asm volatile("s_wait_asynccnt 0x0" ::: "memory");  // this wave's tile landed
    __syncthreads();   // all waves' loads visible; everyone done reading buf^1
    if (base + KNN_TILE < V) issue_tile(base + KNN_TILE, buf ^ 1);
    if (active) {
      for (int j = 0; j < KNN_TILE; ++j) {
        const float x = tile[buf][j * 3 + 0];
        const float y = tile[buf][j * 3 + 1];
        const float z = tile[buf][j * 3 + 2];
        const float d = -2.f * (qx * x + qy * y + qz * z) + qn + (x * x + y * y + z * z);
        if (d < worst) {            // common path: one register compare
          int p = KP1 - 1;
          while (p > 0 && d < bd[p - 1]) { bd[p] = bd[p - 1]; bi[p] = bi[p - 1]; --p; }
          bd[p] = d; bi[p] = base + j;   // ties keep earlier (smaller) index first
          worst = bd[KP1 - 1];
        }
      }
    }
  }
  if (active)
#pragma unroll
    for (int n = 0; n < K; ++n)
      out[((size_t)b * Q + q) * K + n] = bi[n + 1];   // drop self
}

// ---------------------------------------------------------------------------
// operator3d: one wave per vertex; lane = output channel (K0 == 32 == wave).
// disp vectors live in lanes 0..19, broadcast with __shfl.
// fm0[b][v][k] = relu( sum_s  w0w[s*32+k] * max_n relu(disp_n . w0d[:,s*32+k]) )
// ---------------------------------------------------------------------------
__global__ void op3d_kernel(const float* __restrict__ verts, const int* __restrict__ ni,
                            const float* __restrict__ w0d, const float* __restrict__ w0w,
                            float* __restrict__ fm0) {
  const int lane = threadIdx.x & 31;
  const int wave = threadIdx.x >> 5;
  const int v = blockIdx.x * 4 + wave;
  const int b = blockIdx.y;
  const float* P = verts + (size_t)b * N0 * 3;
  const float cx = P[v * 3 + 0], cy = P[v * 3 + 1], cz = P[v * 3 + 2];
  float dx = 0.f, dy = 0.f, dz = 0.f;
  if (lane < NB) {
    const int j = ni[((size_t)b * N0 + v) * NB + lane];
    dx = P[j * 3 + 0] - cx; dy = P[j * 3 + 1] - cy; dz = P[j * 3 + 2] - cz;
  }
  float w0[SUP], w1[SUP], w2[SUP], m[SUP];
#pragma unroll
  for (int s = 0; s < SUP; ++s) {
    const int c = s * K0 + lane;                 // w0d is (3, SUP*K0)
    w0[s] = w0d[c]; w1[s] = w0d[SUP * K0 + c]; w2[s] = w0d[2 * SUP * K0 + c];
    m[s] = 0.f;                                  // max of relu() >= 0
  }
  for (int n = 0; n < NB; ++n) {
    const float ax = __shfl(dx, n, 32), ay = __shfl(dy, n, 32), az = __shfl(dz, n, 32);
#pragma unroll
    for (int s = 0; s < SUP; ++s) {
      const float t = fmaf(ax, w0[s], fmaf(ay, w1[s], az * w2[s]));
      m[s] = fmaxf(m[s], fmaxf(t, 0.f));
    }
  }
  float acc = 0.f;
#pragma unroll
  for (int s = 0; s < SUP; ++s) acc = fmaf(m[s], w0w[s * K0 + lane], acc);
  fm0[((size_t)b * N0 + v) * K0 + lane] = fmaxf(acc, 0.f);   // relu after operator3d
}

// ---------------------------------------------------------------------------
// fp32 GEMM via V_WMMA_F32_16X16X4_F32:  D[M,N] = A[M,K] * B[K,N] + bias[N]
// One wave computes a 16x16 tile; K stepped by 4.  M,N multiples of 16.
// A frag: lane<16 -> K={0,1}, lane>=16 -> K={2,3}.  B frag: vgpr i -> rows {i,i+2}.
// ---------------------------------------------------------------------------
template <int K>
__global__ void gemm_wmma(const float* __restrict__ A, const float* __restrict__ Bw,
                          const float* __restrict__ bias, float* __restrict__ D,
                          int M, int N) {
  const int wave = threadIdx.x >> 5;
  const int lane = threadIdx.x & 31;
  const int tile = blockIdx.x * (blockDim.x >> 5) + wave;
  const int ntn = N >> 4;
  if (tile >= (M >> 4) * ntn) return;              // wave-uniform; EXEC stays full
  const int m0 = (tile / ntn) << 4;
  const int n0 = (tile % ntn) << 4;
  const int half = lane >> 4;
  const int r = lane & 15;

  v8f acc = {};
#pragma unroll
  for (int kk = 0; kk < K / 4; ++kk) {
    const int ka = kk * 4 + half * 2;
    v2f a, bfr;
    const float* ap = A + (size_t)(m0 + r) * K + ka;
    a.x = ap[0];
    a.y = ap[1];
    bfr.x = Bw[(size_t)(ka + 0) * N + n0 + r];
    bfr.y = Bw[(size_t)(ka + 1) * N + n0 + r];
    acc = __builtin_amdgcn_wmma_f32_16x16x4_f32(false, a, false, bfr,
                                                (short)0, acc, false, false);
  }
  const float bv = bias[n0 + r];
#pragma unroll
  for (int i = 0; i < 8; ++i) {
    const int rowm = m0 + i + half * 8;
    D[(size_t)rowm * N + n0 + r] = acc[i] + bv;
  }
}

// ---------------------------------------------------------------------------
// operator_nd aggregate:  out[c] = relu( feat[...,c] +
//     sum_s max_n relu(disp_n . dw[:, s*C+c]) * feat[b, nbr_n, C + s*C + c] )
// One 32-thread block (single wave) per vertex; neighbor disp/idx in LDS.
// ---------------------------------------------------------------------------
template <int VCNT, int C>
__global__ void opnd_kernel(const float* __restrict__ pts, const int* __restrict__ ni,
                            const float* __restrict__ feat, const float* __restrict__ dw,
                            float* __restrict__ outF) {
  const int b = blockIdx.y, v = blockIdx.x, lane = threadIdx.x;
  __shared__ float sd[NB][3];
  __shared__ int si[NB];
  const float* P = pts + (size_t)b * VCNT * 3;
  const float cx = P[v * 3 + 0], cy = P[v * 3 + 1], cz = P[v * 3 + 2];
  if (lane < NB) {
    const int j = ni[((size_t)b * VCNT + v) * NB + lane];
    si[lane] = j;
    sd[lane][0] = P[j * 3 + 0] - cx;
    sd[lane][1] = P[j * 3 + 1] - cy;
    sd[lane][2] = P[j * 3 + 2] - cz;
  }
  __syncthreads();

  constexpr int F = (SUP + 1) * C;    // feat row width
  constexpr int DW = SUP * C;         // dw row width
  const float* frow = feat + ((size_t)b * VCNT + v) * F;

  for (int cc = 0; cc < C / 32; ++cc) {
    const int c = cc * 32 + lane;
    float w0[SUP], w1[SUP], w2[SUP], m[SUP];
#pragma unroll
    for (int s = 0; s < SUP; ++s) {
      const int col = s * C + c;
      w0[s] = dw[col]; w1[s] = dw[DW + col]; w2[s] = dw[2 * DW + col];
      m[s] = -3.0e38f;                 // theta*feat can be negative
    }
    for (int n = 0; n < NB; ++n) {
      const float ax = sd[n][0], ay = sd[n][1], az = sd[n][2];
      const float* fr = feat + ((size_t)b * VCNT + si[n]) * F + C + c;
#pragma unroll
      for (int s = 0; s < SUP; ++s) {
        const float t = fmaxf(fmaf(ax, w0[s], fmaf(ay, w1[s], az * w2[s])), 0.f);
        m[s] = fmaxf(m[s], t * fr[s * C]);
      }
    }
    float acc = 0.f;
#pragma unroll
    for (int s = 0; s < SUP; ++s) acc += m[s];
    outF[((size_t)b * VCNT + v) * C + c] = fmaxf(frow[c] + acc, 0.f);
  }
}

// ---------------------------------------------------------------------------
// Pooling 1 (fused sampling): 256 sampled rows, max over 8 kNN neighbors.
// ---------------------------------------------------------------------------
__global__ void pool1_kernel(const float* __restrict__ verts, const int* __restrict__ ni8,
                             const float* __restrict__ fm1, float* __restrict__ v1,
                             float* __restrict__ fm1p) {
  const int b = blockIdx.y, i = blockIdx.x, c = threadIdx.x;   // c in [0,64)
  const int row = (i * STRIDE1) & (N0 - 1);
  if (c < 3) v1[((size_t)b * N1 + i) * 3 + c] = verts[((size_t)b * N0 + row) * 3 + c];
  float m = -3.0e38f;
#pragma unroll
  for (int n = 0; n < 8; ++n) {
    const int j = ni8[((size_t)b * N1 + i) * 8 + n];
    m = fmaxf(m, fm1[((size_t)b * N0 + j) * C1 + c]);
  }
  fm1p[((size_t)b * N1 + i) * C1 + c] = m;
}

// ---------------------------------------------------------------------------
// Pooling 2 (fused sampling) + final output write.
// ---------------------------------------------------------------------------
__global__ void pool2_kernel(const float* __restrict__ v1, const int* __restrict__ ni16,
                             const float* __restrict__ fm2, float* __restrict__ out_v2,
                             float* __restrict__ out_fm3) {
  const int b = blockIdx.y, i = blockIdx.x, c = threadIdx.x;   // c in [0,256)
  const int row = (i * STRIDE2) & (N1 - 1);
  if (c < 3) out_v2[((size_t)b * N2 + i) * 3 + c] = v1[((size_t)b * N1 + row) * 3 + c];
  float m = -3.0e38f;
#pragma unroll
  for (int n = 0; n < 16; ++n) {
    const int j = ni16[((size_t)b * N2 + i) * 16 + n];
    m = fmaxf(m, fm2[((size_t)b * N1 + j) * C2 + c]);
  }
  out_fm3[((size_t)b * N2 + i) * C2 + c] = m;
}

// ---------------------------------------------------------------------------
extern "C" void kernel_launch(void* const* d_in, const int* in_sizes, int n_in,
                              void* d_out, int out_size, void* d_ws, size_t ws_size,
                              hipStream_t stream) {
  (void)in_sizes; (void)n_in; (void)out_size; (void)ws_size;
  const float* verts = (const float*)d_in[0];
  const float* w0w   = (const float*)d_in[1];   // (1,1,7,32)
  const float* w0d   = (const float*)d_in[2];   // (3,224)
  const float* w1w   = (const float*)d_in[3];   // (32,512)
  const float* w1b   = (const float*)d_in[4];   // (512,)
  const float* w1d   = (const float*)d_in[5];   // (3,448)
  const float* w2w   = (const float*)d_in[6];   // (64,2048)
  const float* w2b   = (const float*)d_in[7];   // (2048,)
  const float* w2d   = (const float*)d_in[8];   // (3,1792)

  char* ws = (char*)d_ws;
  int*   ni20  = (int*)  (ws + OFF_NI20);
  float* fm0   = (float*)(ws + OFF_FM0);
  float* feat1 = (float*)(ws + OFF_FEAT1);
  float* fm1   = (float*)(ws + OFF_FM1);
  int*   ni8   = (int*)  (ws + OFF_NI8);
  float* v1    = (float*)(ws + OFF_V1);
  float* fm1p  = (float*)(ws + OFF_FM1P);
  int*   ni20b = (int*)  (ws + OFF_NI20B);
  float* feat2 = (float*)(ws + OFF_FEAT2);
  float* fm2   = (float*)(ws + OFF_FM2);
  int*   ni16  = (int*)  (ws + OFF_NI16);

  float* out_v2  = (float*)d_out;               // (8,16,3)
  float* out_fm3 = (float*)d_out + BS * N2 * 3; // (8,16,256)

  // Stage 0: kNN(2048 pts, k=20)
  knn_kernel<NB><<<dim3(N0 / KNN_TILE, BS), KNN_TILE, 0, stream>>>(verts, N0, N0, 1, ni20);
  // operator3d -> fm0 (relu'd)
  op3d_kernel<<<dim3(N0 / 4, BS), 128, 0, stream>>>(verts, ni20, w0d, w0w, fm0);
  // feat1 = fm0 @ W1 + b1   (M=16384, K=32, N=512) via WMMA f32
  gemm_wmma<K0><<<dim3((BS * N0 / 16) * ((SUP + 1) * C1 / 16) / 4), 128, 0, stream>>>(
      fm0, w1w, w1b, feat1, BS * N0, (SUP + 1) * C1);
  // operator_nd layer 1 -> fm1 (relu'd)
  opnd_kernel<N0, C1><<<dim3(N0, BS), 32, 0, stream>>>(verts, ni20, feat1, w1d, fm1);

  // Pool 1: kNN(k=8) only at the 256 sampled query rows; max-pool fm1
  knn_kernel<8><<<dim3(N1 / KNN_TILE, BS), KNN_TILE, 0, stream>>>(verts, N0, N1, STRIDE1, ni8);
  pool1_kernel<<<dim3(N1, BS), C1, 0, stream>>>(verts, ni8, fm1, v1, fm1p);

  // Stage 1: kNN(256 pts, k=20)
  knn_kernel<NB><<<dim3(N1 / KNN_TILE, BS), KNN_TILE, 0, stream>>>(v1, N1, N1, 1, ni20b);
  // feat2 = fm1p @ W2 + b2  (M=2048, K=64, N=2048) via WMMA f32
  gemm_wmma<C1><<<dim3((BS * N1 / 16) * ((SUP + 1) * C2 / 16) / 4), 128, 0, stream>>>(
      fm1p, w2w, w2b, feat2, BS * N1, (SUP + 1) * C2);
  // operator_nd layer 2 -> fm2 (relu'd)
  opnd_kernel<N1, C2><<<dim3(N1, BS), 32, 0, stream>>>(v1, ni20b, feat2, w2d, fm2);

  // Pool 2: kNN(k=16) only at the 16 sampled rows; max-pool fm2; write outputs
  knn_kernel<16><<<dim3(1, BS), KNN_TILE, 0, stream>>>(v1, N1, N2, STRIDE2, ni16);
  pool2_kernel<<<dim3(N2, BS), C2, 0, stream>>>(v1, ni16, fm2, out_v2, out_fm3);
}